// UnrolledSinkhorn_22067541967475
// MI455X (gfx1250) — compile-verified
//
#include <hip/hip_runtime.h>
#include <math.h>

// Sinkhorn unrolled scan on gfx1250.
// EPS=1e-4 amplifies distance error 1e4x inside the exponent -> A/B must stay
// fp32 -> V_WMMA_F32_16X16X4_F32. Flash (online) softmax per 16-wide column
// tile with double-buffered B fragments so global_load_b64 clauses overlap the
// WMMA chain instead of serializing with it. Both scan directions fused into
// one launch (blockIdx.z) since same-stream kernels serialize.

#define EPS_F         1.0e-4f
#define INV_EPS_F     1.0e4f
#define TWO_INV_EPS_F 2.0e4f
#define D_DIM 64
#define NDIM  2048
#define BATCH 4
#define TSTEPS 16

typedef float v2f __attribute__((ext_vector_type(2)));
typedef float v8f __attribute__((ext_vector_type(8)));

__device__ __forceinline__ float wshfl(float v, int src)      { return __shfl(v, src, 32); }
__device__ __forceinline__ float wshfl_xor(float v, int mask) { return __shfl_xor(v, mask, 32); }

// Issue one clause of 16 b64 loads for the B tile at column base j0 (+ colc).
// y over-read past NDIM is safe (stays inside the (b,t,n,d) allocation for all
// reachable slices); colc index is clamped.
__device__ __forceinline__ void load_btile(const float* __restrict__ Yb,
                                           const float* __restrict__ cc,
                                           int j0, int l16, int h,
                                           v2f (&b)[16], float& cN) {
  const int N = j0 + l16;
  const float* yr = Yb + (long)N * D_DIM + 2 * h;
#pragma unroll
  for (int kc = 0; kc < 16; ++kc) b[kc] = *(const v2f*)(yr + 4 * kc);
  const int Nc = (N < NDIM) ? N : (NDIM - 1);
  cN = cc[Nc];
}

// 16-step K chain of V_WMMA_F32_16X16X4_F32 + single-exp online LSE update.
__device__ __forceinline__ void wmma_tile(const v2f (&a)[16], const v2f (&b)[16],
                                          const float (&rowc)[8], float colN,
                                          float (&mrow)[8], float (&srow)[8]) {
  v8f acc = {0.f, 0.f, 0.f, 0.f, 0.f, 0.f, 0.f, 0.f};
#pragma unroll
  for (int kc = 0; kc < 16; ++kc) {
    // (neg_a, A, neg_b, B, c_mod, C, reuse_a, reuse_b)
    acc = __builtin_amdgcn_wmma_f32_16x16x4_f32(
        false, a[kc], false, b[kc], (short)0, acc, false, false);
  }
#pragma unroll
  for (int v = 0; v < 8; ++v) {
    float s = fmaf(TWO_INV_EPS_F, acc[v], rowc[v] + colN);
    float e = __expf(-fabsf(s - mrow[v]));          // exp(min-max); 1st tile: exp(-inf)=0
    srow[v] = (s > mrow[v]) ? fmaf(srow[v], e, 1.f) : (srow[v] + e);
    mrow[v] = fmaxf(mrow[v], s);
  }
}

// colc[dir][b*NDIM + j] = (carry[b,j] - ||col_j||^2)/EPS + lw[b,j]; dir via blockIdx.y
__global__ __launch_bounds__(256) void colconst_kernel(
    const float* __restrict__ Yc0, const float* __restrict__ C0, const float* __restrict__ L0,
    const float* __restrict__ Yc1, const float* __restrict__ C1, const float* __restrict__ L1,
    float* __restrict__ colcF, float* __restrict__ colcG,
    long bstrideX, long bstrideC)
{
  const int dir = blockIdx.y;
  const float* Ycols = dir ? Yc1 : Yc0;
  const float* CARRY = dir ? C1  : C0;
  const float* LW    = dir ? L1  : L0;
  float*       colc  = dir ? colcG : colcF;

  int idx = blockIdx.x * blockDim.x + threadIdx.x;   // over BATCH*NDIM
  int bb = idx / NDIM, j = idx % NDIM;
  const float4* row = (const float4*)(Ycols + (long)bb * bstrideX + (long)j * D_DIM);
  float nn = 0.f;
#pragma unroll
  for (int q = 0; q < D_DIM / 4; ++q) {
    float4 t = row[q];
    nn += t.x * t.x + t.y * t.y + t.z * t.z + t.w * t.w;
  }
  long o = (long)bb * bstrideC + j;
  colc[idx] = (CARRY[o] - nn) * INV_EPS_F + LW[o];
}

// out[b,i] = -EPS * logsumexp_j( 2e4*(x_i . y_j) + rowc[i] + colc[j] )
// One wave -> 16 rows. blockIdx.z selects direction (f vs g).
__global__ __launch_bounds__(256) void lse_rows_kernel(
    const float* __restrict__ X0, const float* __restrict__ Y0,
    const float* __restrict__ cc0, float* __restrict__ O0,
    const float* __restrict__ X1, const float* __restrict__ Y1,
    const float* __restrict__ cc1, float* __restrict__ O1,
    long bstrideX, long bstrideC)
{
  const int dir  = blockIdx.z;
  const float* X    = dir ? X1 : X0;
  const float* Y    = dir ? Y1 : Y0;
  const float* colc = dir ? cc1 : cc0;
  float*       OUT  = dir ? O1 : O0;

  const int bb   = blockIdx.y;
  const int wave = threadIdx.x >> 5;
  const int lane = threadIdx.x & 31;
  const int h    = lane >> 4;   // half of the wave: selects K/M offset
  const int l16  = lane & 15;
  const int r0   = blockIdx.x * 128 + wave * 16;   // 8 waves * 16 rows per block

  const float* Xb = X + (long)bb * bstrideX;
  const float* Yb = Y + (long)bb * bstrideX;
  const float* cc = colc + (long)bb * NDIM;
  float*       op = OUT + (long)bb * bstrideC;

  // A fragments: 16 chunks of 16x4 f32 (K=64). Lane l16 = row M=l16, pair
  // {K=4kc+2h, 4kc+2h+1} per the 32-bit 16x4 A layout (ISA 7.12.2).
  v2f a[16];
  const float* xrow = Xb + (long)(r0 + l16) * D_DIM + 2 * h;
  float xpart = 0.f;
#pragma unroll
  for (int kc = 0; kc < 16; ++kc) {
    v2f t = *(const v2f*)(xrow + 4 * kc);
    a[kc] = t;
    xpart += t.x * t.x + t.y * t.y;
  }
  float xfull = xpart + wshfl_xor(xpart, 16);       // ||x_{l16}||^2 on both half-lanes
  float rowc[8];
#pragma unroll
  for (int v = 0; v < 8; ++v) rowc[v] = -INV_EPS_F * wshfl(xfull, 8 * h + v);

  float mrow[8], srow[8];
#pragma unroll
  for (int v = 0; v < 8; ++v) { mrow[v] = -INFINITY; srow[v] = 0.f; }

  // Double-buffered column-tile loop: prefetch tile t+1, compute tile t.
  v2f b0[16], b1[16];
  float c0, c1;
  load_btile(Yb, cc, 0, l16, h, b0, c0);
  for (int j0 = 0; j0 < NDIM; j0 += 32) {
    load_btile(Yb, cc, j0 + 16, l16, h, b1, c1);    // prefetch tile t+1
    wmma_tile(a, b0, rowc, c0, mrow, srow);         // compute tile t
    load_btile(Yb, cc, j0 + 32, l16, h, b0, c0);    // prefetch tile t+2 (tail over-read safe)
    wmma_tile(a, b1, rowc, c1, mrow, srow);         // compute tile t+1
  }

  // Merge across the 16 lanes of each half (same rows, different columns).
#pragma unroll
  for (int v = 0; v < 8; ++v) {
    float m = mrow[v], s = srow[v];
#pragma unroll
    for (int off = 1; off < 16; off <<= 1) {
      float om = wshfl_xor(m, off);
      float os = wshfl_xor(s, off);
      float nm = fmaxf(m, om);
      s = s * __expf(m - nm) + os * __expf(om - nm);
      m = nm;
    }
    mrow[v] = m; srow[v] = s;
  }
  if (l16 == 0) {
#pragma unroll
    for (int v = 0; v < 8; ++v)
      op[r0 + 8 * h + v] = -EPS_F * (mrow[v] + logf(srow[v]));
  }
}

// loss = sum_{b,i} exp(la_new[b,i]) * (f_new[b,i] + g_new[b,i])   (n==m)
__global__ __launch_bounds__(256) void loss_kernel(
    const float* __restrict__ F, const float* __restrict__ G,
    const float* __restrict__ LA, float* __restrict__ acc, long bstrideC)
{
  int idx = blockIdx.x * blockDim.x + threadIdx.x;   // over BATCH*NDIM
  int bb = idx / NDIM;
  int i  = idx % NDIM;
  long o = (long)bb * bstrideC + i;
  float w = __expf(LA[o]) * (F[o] + G[o]);
  for (int off = 1; off < 32; off <<= 1) w += __shfl_xor(w, off, 32);
  __shared__ float red[8];
  if ((threadIdx.x & 31) == 0) red[threadIdx.x >> 5] = w;
  __syncthreads();
  if (threadIdx.x < 8) {
    float v = red[threadIdx.x];
    for (int off = 1; off < 8; off <<= 1) v += __shfl_xor(v, off, 32);
    if (threadIdx.x == 0) atomicAdd(acc, v);
  }
}

__global__ void init_kernel(float* f0, float* g0, float* accs, long bstrideC) {
  int idx = blockIdx.x * blockDim.x + threadIdx.x;
  if (idx < BATCH * NDIM) {
    int bb = idx / NDIM, i = idx % NDIM;
    f0[(long)bb * bstrideC + i] = 0.f;
    g0[(long)bb * bstrideC + i] = 0.f;
  }
  if (idx < (TSTEPS - 1)) accs[idx] = 0.f;
}

__global__ void finalize_losses(float* losses, const float* accs) {
  int idx = threadIdx.x;
  if (idx < BATCH * TSTEPS) {
    int tt = idx % TSTEPS;
    losses[idx] = (tt == 0) ? 0.f : accs[tt - 1];
  }
}

extern "C" void kernel_launch(void* const* d_in, const int* in_sizes, int n_in,
                              void* d_out, int out_size, void* d_ws, size_t ws_size,
                              hipStream_t stream) {
  const float* x    = (const float*)d_in[0];   // (4,16,2048,64)
  const float* y    = (const float*)d_in[1];   // (4,16,2048,64)
  const float* loga = (const float*)d_in[2];   // (4,16,2048)
  const float* logb = (const float*)d_in[3];   // (4,16,2048)

  float* losses = (float*)d_out;                              // 64
  float* f      = losses + BATCH * TSTEPS;                    // 4*16*2048
  float* g      = f + (long)BATCH * TSTEPS * NDIM;            // 4*16*2048

  float* accs  = (float*)d_ws;                                // 15 floats (+pad)
  float* colcF = accs + 16;                                   // BATCH*NDIM
  float* colcG = colcF + BATCH * NDIM;                        // BATCH*NDIM

  const long bstrideX = (long)TSTEPS * NDIM * D_DIM;
  const long bstrideC = (long)TSTEPS * NDIM;

  init_kernel<<<(BATCH * NDIM + 255) / 256, 256, 0, stream>>>(f, g, accs, bstrideC);

  dim3 block(256);
  dim3 gridLse(NDIM / 128, BATCH, 2);
  dim3 gridCC((BATCH * NDIM) / 256, 2);

  for (int k = 1; k < TSTEPS; ++k) {
    const float* xk    = x + (long)k       * NDIM * D_DIM;
    const float* ykm1  = y + (long)(k - 1) * NDIM * D_DIM;
    const float* yk    = y + (long)k       * NDIM * D_DIM;
    const float* xkm1  = x + (long)(k - 1) * NDIM * D_DIM;
    const float* gprev = g + (long)(k - 1) * NDIM;
    const float* fprev = f + (long)(k - 1) * NDIM;
    const float* lbp   = logb + (long)(k - 1) * NDIM;
    const float* lap   = loga + (long)(k - 1) * NDIM;
    float* fnew = f + (long)k * NDIM;
    float* gnew = g + (long)k * NDIM;

    // Fold per-column constants for both directions in one launch.
    colconst_kernel<<<gridCC, block, 0, stream>>>(
        ykm1, gprev, lbp, xkm1, fprev, lap, colcF, colcG, bstrideX, bstrideC);

    // Both directions in one launch: z=0 -> f_new (x_t vs y_{t-1}),
    //                                z=1 -> g_new (y_t vs x_{t-1}).
    lse_rows_kernel<<<gridLse, block, 0, stream>>>(
        xk, ykm1, colcF, fnew, yk, xkm1, colcG, gnew, bstrideX, bstrideC);

    loss_kernel<<<(BATCH * NDIM) / 256, 256, 0, stream>>>(
        fnew, gnew, loga + (long)k * NDIM, accs + (k - 1), bstrideC);
  }
  finalize_losses<<<1, 64, 0, stream>>>(losses, accs);
}